// ICPModule_50276887167686
// MI455X (gfx1250) — compile-verified
//
#include <hip/hip_runtime.h>

typedef float v2f __attribute__((ext_vector_type(2)));
typedef float v8f __attribute__((ext_vector_type(8)));

#define B_   8
#define N_   2048
#define M_   2048
#define THR2 0.25f   // THRESHOLD^2

// d_ws float layout:
//   [0, 128)            : T matrices, B x (4x4 row-major)
//   [128, 128 + 16384)  : per-wave partial sums, (B * N_/16) waves x 16 floats

// ---------------------------------------------------------------------------
// Kernel A: nearest-neighbor search via V_WMMA_F32_16X16X4_F32 + ICP partial
// sums. One wave handles a 16-point source tile. Targets are staged in LDS
// pre-swizzled into the exact per-lane WMMA A-operand layout:
//   lane l (0-15):  [t_x, t_y]   (K=0,1 of row l)
//   lane 16+l:      [t_z, |t|^2] (K=2,3 of row l)
// Inner loop: prefetched ds_load_b64 -> v_wmma -> v_min3 tree -> rare update.
// ---------------------------------------------------------------------------
__global__ __launch_bounds__(256) void icp_nn_kernel(
    const float* __restrict__ src, const float* __restrict__ tgt,
    const float* __restrict__ Tmat, float* __restrict__ partials) {
  __shared__ v2f sT2[2 * M_];  // 32 KB, indexed [tile*32 + lane]

  const int b        = blockIdx.x >> 4;        // 16 blocks per batch
  const int tileBase = (blockIdx.x & 15) * 8;  // 8 waves -> 8 source tiles

  // Stage targets for this batch into LDS (pre-swizzled, norms precomputed).
  const float* tb = tgt + (size_t)b * M_ * 3;
  for (int i = threadIdx.x; i < M_; i += 256) {
    float tx = tb[i * 3 + 0], ty = tb[i * 3 + 1], tz = tb[i * 3 + 2];
    float tn = tx * tx + ty * ty + tz * tz;
    int mt = i >> 4, l = i & 15;
    sT2[(mt << 5) + l]      = (v2f){tx, ty};
    sT2[(mt << 5) + 16 + l] = (v2f){tz, tn};
  }
  __syncthreads();

  const int wave = threadIdx.x >> 5;
  const int lane = threadIdx.x & 31;
  const int s    = lane & 15;   // source index within tile
  const int half = lane >> 4;   // 0: rows 0-7, 1: rows 8-15 of each D tile
  const int srcTile = tileBase + wave;

  // Current transform for this batch (uniform).
  const float* Tb = Tmat + b * 16;
  const float R00 = Tb[0], R01 = Tb[1], R02 = Tb[2],  t0 = Tb[3];
  const float R10 = Tb[4], R11 = Tb[5], R12 = Tb[6],  t1 = Tb[7];
  const float R20 = Tb[8], R21 = Tb[9], R22 = Tb[10], t2 = Tb[11];

  // Load + transform source point (both wave halves load the same point).
  const float* sp = src + ((size_t)b * N_ + srcTile * 16 + s) * 3;
  const float sx = sp[0], sy = sp[1], sz = sp[2];
  const float px = R00 * sx + R01 * sy + R02 * sz + t0;
  const float py = R10 * sx + R11 * sy + R12 * sz + t1;
  const float pz = R20 * sx + R21 * sy + R22 * sz + t2;
  const float pn = px * px + py * py + pz * pz;

  // B operand column s = [-2px, -2py, -2pz, 1]  (4x16, K striped over halves)
  v2f bop;
  bop.x = half ? (-2.0f * pz) : (-2.0f * px);
  bop.y = half ? 1.0f         : (-2.0f * py);

  float minv = 3.4e38f;
  int   mini = 0;
  const int rowOff = half * 8;

  // Register double-buffer: issue next tile's LDS load before consuming
  // the current one so ds latency hides under the WMMA + min tree.
  v2f aopCur = sT2[lane];
  #pragma unroll 4
  for (int mt = 0; mt < M_ / 16; ++mt) {
    v2f aopNext = sT2[(((mt + 1) & 127) << 5) + lane];

    v8f c = {};
    // D[t][s] = -2 * dot(s_xformed, t) + |t|^2   (argmin key; +|s|^2 later)
    c = __builtin_amdgcn_wmma_f32_16x16x4_f32(
        false, aopCur, false, bop, (short)0, c, false, false);

    // Tile minimum via independent min tree (v_min3), depth 3.
    float m = fminf(fminf(fminf(c[0], c[1]), fminf(c[2], c[3])),
                    fminf(fminf(c[4], c[5]), fminf(c[6], c[7])));

    // Rare slow path: recover row index only when this tile wins.
    if (m < minv) {
      minv = m;
      int r = 7;
      #pragma unroll
      for (int k = 6; k >= 0; --k)
        if (c[k] == m) r = k;  // descending scan -> smallest matching row
      mini = mt * 16 + rowOff + r;
    }
    aopCur = aopNext;
  }

  // Combine the two wave halves (rows 0-7 vs rows 8-15 of every tile).
  {
    float ov = __shfl_xor(minv, 16);
    int   oi = __shfl_xor(mini, 16);
    if (ov < minv) { minv = ov; mini = oi; }
  }

  const float mind2 = minv + pn;
  float w = (mind2 < THR2) ? 1.0f : 0.0f;
  if (half) w = 0.0f;  // lanes 16-31 are duplicates; avoid double counting

  // Gather nearest neighbor.
  const float* np = tgt + ((size_t)b * M_ + mini) * 3;
  const float nx = np[0], ny = np[1], nz = np[2];

  // 16 ICP sums: Sw, Sw*s(3), Sw*t(3), Sw*s t^T (9)
  float vals[16];
  vals[0]  = w;
  vals[1]  = w * px;      vals[2]  = w * py;      vals[3]  = w * pz;
  vals[4]  = w * nx;      vals[5]  = w * ny;      vals[6]  = w * nz;
  vals[7]  = w * px * nx; vals[8]  = w * px * ny; vals[9]  = w * px * nz;
  vals[10] = w * py * nx; vals[11] = w * py * ny; vals[12] = w * py * nz;
  vals[13] = w * pz * nx; vals[14] = w * pz * ny; vals[15] = w * pz * nz;

  #pragma unroll
  for (int j = 0; j < 16; ++j) {
    float v = vals[j];
    #pragma unroll
    for (int off = 16; off >= 1; off >>= 1) v += __shfl_xor(v, off);
    vals[j] = v;
  }

  if (lane == 0) {
    float* out = partials + ((size_t)b * (N_ / 16) + srcTile) * 16;
    #pragma unroll
    for (int j = 0; j < 16; ++j) out[j] = vals[j];
  }
}

// ---------------------------------------------------------------------------
// Kernel B: per-batch reduction, 3x3 SVD (Jacobi on H^T H), transform update.
// One lane per batch; deterministic serial summation.
// ---------------------------------------------------------------------------
__global__ __launch_bounds__(32) void icp_update_kernel(
    const float* __restrict__ partials, float* __restrict__ Tmat,
    float* __restrict__ out) {
  const int b = threadIdx.x;
  if (b >= B_) return;

  float sums[16];
  #pragma unroll
  for (int j = 0; j < 16; ++j) sums[j] = 0.0f;
  const float* p = partials + (size_t)b * (N_ / 16) * 16;
  for (int wv = 0; wv < N_ / 16; ++wv)
    #pragma unroll
    for (int j = 0; j < 16; ++j) sums[j] += p[wv * 16 + j];

  const float Sw = sums[0];
  const float Ss[3] = {sums[1], sums[2], sums[3]};
  const float St[3] = {sums[4], sums[5], sums[6]};
  const float wsum = fmaxf(Sw, 1e-6f);
  float cs[3], ct[3];
  for (int i = 0; i < 3; ++i) { cs[i] = Ss[i] / wsum; ct[i] = St[i] / wsum; }

  // H = sum w*s*t^T - Ss*ct^T - cs*St^T + Sw*cs*ct^T
  float H[3][3];
  for (int i = 0; i < 3; ++i)
    for (int j = 0; j < 3; ++j)
      H[i][j] = sums[7 + i * 3 + j] - ct[j] * Ss[i] - cs[i] * St[j] +
                Sw * cs[i] * ct[j];

  // A = H^T H (symmetric), Jacobi eigendecomposition -> V (right sing. vecs)
  float A[3][3];
  for (int i = 0; i < 3; ++i)
    for (int j = 0; j < 3; ++j) {
      float acc = 0.0f;
      for (int k = 0; k < 3; ++k) acc += H[k][i] * H[k][j];
      A[i][j] = acc;
    }
  float V[3][3] = {{1, 0, 0}, {0, 1, 0}, {0, 0, 1}};
  for (int sweep = 0; sweep < 12; ++sweep) {
    for (int pi = 0; pi < 3; ++pi)
      for (int qi = pi + 1; qi < 3; ++qi) {
        float apq = A[pi][qi];
        if (fabsf(apq) > 1e-20f) {
          float tau = (A[qi][qi] - A[pi][pi]) / (2.0f * apq);
          float t = (tau >= 0.0f ? 1.0f : -1.0f) /
                    (fabsf(tau) + sqrtf(1.0f + tau * tau));
          float cth = 1.0f / sqrtf(1.0f + t * t);
          float sth = t * cth;
          for (int k = 0; k < 3; ++k) {
            float akp = A[k][pi], akq = A[k][qi];
            A[k][pi] = cth * akp - sth * akq;
            A[k][qi] = sth * akp + cth * akq;
          }
          for (int k = 0; k < 3; ++k) {
            float apk = A[pi][k], aqk = A[qi][k];
            A[pi][k] = cth * apk - sth * aqk;
            A[qi][k] = sth * apk + cth * aqk;
          }
          for (int k = 0; k < 3; ++k) {
            float vkp = V[k][pi], vkq = V[k][qi];
            V[k][pi] = cth * vkp - sth * vkq;
            V[k][qi] = sth * vkp + cth * vkq;
          }
        }
      }
  }
  float eig[3] = {A[0][0], A[1][1], A[2][2]};
  // sort eigenvalues descending, permuting V columns
  for (int i = 0; i < 2; ++i)
    for (int j = 0; j < 2 - i; ++j)
      if (eig[j] < eig[j + 1]) {
        float e = eig[j]; eig[j] = eig[j + 1]; eig[j + 1] = e;
        for (int k = 0; k < 3; ++k) {
          float v = V[k][j]; V[k][j] = V[k][j + 1]; V[k][j + 1] = v;
        }
      }

  // U columns: u_i = H v_i / sigma_i (i=0,1); u_2 = u_0 x u_1
  float U[3][3];
  for (int cidx = 0; cidx < 2; ++cidx) {
    float sig = sqrtf(fmaxf(eig[cidx], 0.0f));
    float vx = V[0][cidx], vy = V[1][cidx], vz = V[2][cidx];
    float ux = H[0][0] * vx + H[0][1] * vy + H[0][2] * vz;
    float uy = H[1][0] * vx + H[1][1] * vy + H[1][2] * vz;
    float uz = H[2][0] * vx + H[2][1] * vy + H[2][2] * vz;
    float inv = (sig > 1e-12f) ? 1.0f / sig : 0.0f;
    U[0][cidx] = ux * inv; U[1][cidx] = uy * inv; U[2][cidx] = uz * inv;
  }
  U[0][2] = U[1][0] * U[2][1] - U[2][0] * U[1][1];
  U[1][2] = U[2][0] * U[0][1] - U[0][0] * U[2][1];
  U[2][2] = U[0][0] * U[1][1] - U[1][0] * U[0][1];

  // d = sign(det(V U^T)) = sign(det V * det U)
  float detV = V[0][0] * (V[1][1] * V[2][2] - V[1][2] * V[2][1]) -
               V[0][1] * (V[1][0] * V[2][2] - V[1][2] * V[2][0]) +
               V[0][2] * (V[1][0] * V[2][1] - V[1][1] * V[2][0]);
  float detU = U[0][0] * (U[1][1] * U[2][2] - U[1][2] * U[2][1]) -
               U[0][1] * (U[1][0] * U[2][2] - U[1][2] * U[2][0]) +
               U[0][2] * (U[1][0] * U[2][1] - U[1][1] * U[2][0]);
  float dd = detV * detU;
  float d = (dd > 0.0f) ? 1.0f : ((dd < 0.0f) ? -1.0f : 0.0f);
  float scl[3] = {1.0f, 1.0f, d};

  float Rd[3][3];
  for (int i = 0; i < 3; ++i)
    for (int j = 0; j < 3; ++j) {
      float acc = 0.0f;
      for (int k = 0; k < 3; ++k) acc += V[i][k] * scl[k] * U[j][k];
      Rd[i][j] = acc;
    }
  float td[3];
  for (int i = 0; i < 3; ++i)
    td[i] = ct[i] - (Rd[i][0] * cs[0] + Rd[i][1] * cs[1] + Rd[i][2] * cs[2]);

  // T = delta @ T
  float* Tb = Tmat + b * 16;
  float Told[16];
  #pragma unroll
  for (int k = 0; k < 16; ++k) Told[k] = Tb[k];
  float delta[16] = {Rd[0][0], Rd[0][1], Rd[0][2], td[0],
                     Rd[1][0], Rd[1][1], Rd[1][2], td[1],
                     Rd[2][0], Rd[2][1], Rd[2][2], td[2],
                     0.0f,     0.0f,     0.0f,     1.0f};
  float Tn[16];
  for (int r = 0; r < 4; ++r)
    for (int c = 0; c < 4; ++c) {
      float acc = 0.0f;
      for (int k = 0; k < 4; ++k) acc += delta[r * 4 + k] * Told[k * 4 + c];
      Tn[r * 4 + c] = acc;
    }
  #pragma unroll
  for (int k = 0; k < 16; ++k) { Tb[k] = Tn[k]; out[b * 16 + k] = Tn[k]; }
}

__global__ void icp_init_kernel(float* __restrict__ Tmat) {
  int i = threadIdx.x;
  if (i < B_ * 16) {
    int r = (i >> 2) & 3, c = i & 3;
    Tmat[i] = (r == c) ? 1.0f : 0.0f;
  }
}

extern "C" void kernel_launch(void* const* d_in, const int* in_sizes, int n_in,
                              void* d_out, int out_size, void* d_ws,
                              size_t ws_size, hipStream_t stream) {
  const float* src = (const float*)d_in[0];  // (B, N, 3)
  const float* tgt = (const float*)d_in[1];  // (B, M, 3)
  float* out = (float*)d_out;                // (B, 4, 4)
  float* ws = (float*)d_ws;
  float* Tmat = ws;              // 128 floats
  float* partials = ws + 128;    // B * 128 waves * 16 floats = 16384 floats

  hipLaunchKernelGGL(icp_init_kernel, dim3(1), dim3(128), 0, stream, Tmat);
  for (int it = 0; it < 50; ++it) {
    hipLaunchKernelGGL(icp_nn_kernel, dim3(B_ * 16), dim3(256), 0, stream,
                       src, tgt, Tmat, partials);
    hipLaunchKernelGGL(icp_update_kernel, dim3(1), dim3(32), 0, stream,
                       partials, Tmat, out);
  }
}